// Qwen3_5MoeSparseMoeBlock_58600533787444
// MI455X (gfx1250) — compile-verified
//
#include <hip/hip_runtime.h>
#include <math.h>

typedef float v2f __attribute__((ext_vector_type(2)));
typedef float v4f __attribute__((ext_vector_type(4)));
typedef float v8f __attribute__((ext_vector_type(8)));

#define T_TOK 2048
#define HDIM  1024
#define NEXP  64
#define TOPK  8
#define IDIM  512            // expert intermediate
#define ISH   2048           // shared intermediate
#define CAPE  2048           // per-expert max tokens
#define RPAD_MAX (T_TOK*TOPK + NEXP*32)   // 18432 >= worst padded row total (18368)
#define ROW_TILES (RPAD_MAX / 32)         // 576
#define LDA  68              // A tile stride (64 + 4 pad)
#define LDBI 264             // interleaved B tile stride: 2*128 cols + 8 pad (even, 16B-friendly)

__device__ __forceinline__ float sig_(float x)  { return 1.0f / (1.0f + __expf(-x)); }
__device__ __forceinline__ float silu_(float x) { return x * sig_(x); }

#define WMMA_F32(a, b, c) \
    __builtin_amdgcn_wmma_f32_16x16x4_f32(false, (a), false, (b), (short)0, (c), false, false)

// ---------------------------------------------------------------- router
__global__ void router_kernel(const float* __restrict__ x,
                              const float* __restrict__ gate_w,      // [H,E]
                              const float* __restrict__ sgw,         // [H,1]
                              float* __restrict__ token_gate,        // [T]
                              int*   __restrict__ counts,            // [E]
                              int*   __restrict__ tok_list,          // [E*CAPE]
                              float* __restrict__ w_list)            // [E*CAPE]
{
    __shared__ float xs[HDIM];
    __shared__ float pr[NEXP];
    __shared__ float red[64];
    const int t   = blockIdx.x;
    const int tid = threadIdx.x;          // 0..63
    const float* xrow = x + (size_t)t * HDIM;
    for (int h = tid; h < HDIM; h += 64) xs[h] = xrow[h];
    __syncthreads();

    float acc = 0.f;
    for (int h = 0; h < HDIM; ++h)
        acc += xs[h] * gate_w[(size_t)h * NEXP + tid];
    pr[tid] = acc;

    float g = 0.f;
    for (int h = tid; h < HDIM; h += 64) g += xs[h] * sgw[h];
    red[tid] = g;
    __syncthreads();

    if (tid == 0) {
        float s = 0.f;
        for (int i = 0; i < 64; ++i) s += red[i];
        token_gate[t] = sig_(s);

        float mx = pr[0];
        for (int e = 1; e < NEXP; ++e) mx = fmaxf(mx, pr[e]);
        for (int e = 0; e < NEXP; ++e) pr[e] = __expf(pr[e] - mx);

        int   sel[TOPK];
        float sw[TOPK];
        float ssum = 0.f;
        for (int k = 0; k < TOPK; ++k) {
            int bi = 0; float bv = -1.f;
            for (int e = 0; e < NEXP; ++e)
                if (pr[e] > bv) { bv = pr[e]; bi = e; }
            sel[k] = bi; sw[k] = bv; ssum += bv; pr[bi] = -2.f;
        }
        const float inv = 1.f / ssum;
        for (int k = 0; k < TOPK; ++k) {
            const int e    = sel[k];
            const int slot = atomicAdd(&counts[e], 1);
            tok_list[e * CAPE + slot] = t;
            w_list [e * CAPE + slot] = sw[k] * inv;
        }
    }
}

// ---------------------------------------------------------------- padded prefix (pad 32)
__global__ void offsets_kernel(const int* __restrict__ counts, int* __restrict__ off)
{
    if (threadIdx.x == 0) {
        int acc = 0;
        for (int e = 0; e < NEXP; ++e) { off[e] = acc; acc += (counts[e] + 31) & ~31; }
        off[NEXP] = acc;
    }
}

// B-panel stage, k-pair interleaved: Bs[p*LDBI + 2c + (k&1)] = B[kb+2p+(k&1)][c]
// 64 k-rows x 128 cols from two 64-col column groups p0/p1.
// Each thread: two v2f global loads (rows k,k+1; cols c,c+1), one b128 LDS store.
__device__ __forceinline__ void stage_B(float* Bs, const float* p0, const float* p1,
                                        int kb, int ldb, int tid)
{
    const int cfix = (tid & 63) * 2;                    // 0..126, wave-uniform half
    const float* src = (cfix < 64) ? (p0 + cfix) : (p1 + (cfix - 64));
    float* dstbase = Bs + 2 * cfix;
    #pragma unroll
    for (int j = 0; j < 8; ++j) {
        const int p = (tid >> 6) + 4 * j;               // k-pair index 0..31
        const int k = kb + 2 * p;
        v2f lo = *(const v2f*)(src + (size_t)k * ldb);
        v2f hi = *(const v2f*)(src + (size_t)(k + 1) * ldb);
        v4f v; v.x = lo.x; v.y = hi.x; v.z = lo.y; v.w = hi.y;
        *(v4f*)(dstbase + p * LDBI) = v;
    }
}

// fragment fetch: pair p = kk/2 + hs; column n; {B[k][n],B[k+1][n]} contiguous b64
#define B_FRAG(Bs, kk, hs, n) (*(const v2f*)((Bs) + ((kk >> 1) + (hs)) * LDBI + 2 * (n)))

// ---------------------------------------------------------------- shared gate-up + SwiGLU
__global__ void shared_gateup_kernel(const float* __restrict__ x,     // [T,H]
                                     const float* __restrict__ w,     // [H, 2*ISH]
                                     float* __restrict__ act)         // [T, ISH]
{
    __shared__ float As[32 * LDA];
    __shared__ float Bs[32 * LDBI];
    const int tid  = threadIdx.x;           // 0..255
    const int lane = tid & 31;
    const int wv   = tid >> 5;              // 0..7
    const int mrow = wv >> 2;               // 0..1
    const int ncol = wv & 3;                // 0..3
    const int hs   = lane >> 4;
    const int lr   = lane & 15;
    const int r0   = blockIdx.y * 32;
    const int c0g  = blockIdx.x * 64;       // intermediate col group base
    const int ldb  = 2 * ISH;
    const int nb   = 16 * ncol + lr;

    const float* p0 = w + c0g;              // gate cols
    const float* p1 = w + ISH + c0g;        // up cols
    v8f cg = {}; v8f cu = {};

    for (int kb = 0; kb < HDIM; kb += 64) {
        __syncthreads();
        {
            const int row = tid >> 3, col = (tid & 7) * 8;
            const float* src = x + (size_t)(r0 + row) * HDIM + kb + col;
            float* dst = As + row * LDA + col;
            *(v4f*)(dst)     = *(const v4f*)(src);
            *(v4f*)(dst + 4) = *(const v4f*)(src + 4);
        }
        stage_B(Bs, p0, p1, kb, ldb, tid);
        __syncthreads();
        #pragma unroll
        for (int kk = 0; kk < 64; kk += 4) {
            v2f a = *(const v2f*)(As + (16 * mrow + lr) * LDA + kk + 2 * hs);
            v2f b0 = B_FRAG(Bs, kk, hs, nb);
            cg = WMMA_F32(a, b0, cg);
            v2f b1 = B_FRAG(Bs, kk, hs, nb + 64);
            cu = WMMA_F32(a, b1, cu);
        }
    }
    #pragma unroll
    for (int g = 0; g < 8; ++g) {
        const int row  = r0 + 16 * mrow + g + 8 * hs;
        act[(size_t)row * ISH + c0g + nb] = silu_(cg[g]) * cu[g];
    }
}

// ---------------------------------------------------------------- shared down * token gate
__global__ void shared_down_kernel(const float* __restrict__ act,   // [T, ISH]
                                   const float* __restrict__ w,     // [ISH, H]
                                   const float* __restrict__ tg,    // [T]
                                   float* __restrict__ out)         // [T, H]
{
    __shared__ float As[32 * LDA];
    __shared__ float Bs[32 * LDBI];
    const int tid  = threadIdx.x;
    const int lane = tid & 31;
    const int wv   = tid >> 5;
    const int mrow = wv >> 2;
    const int ncol = wv & 3;
    const int hs   = lane >> 4;
    const int lr   = lane & 15;
    const int r0   = blockIdx.y * 32;
    const int c0   = blockIdx.x * 128;      // output col group base
    const int ldb  = HDIM;
    const int nb   = 16 * ncol + lr;

    const float* p0 = w + c0;
    const float* p1 = w + c0 + 64;
    v8f ca = {}; v8f cb = {};

    for (int kb = 0; kb < ISH; kb += 64) {
        __syncthreads();
        {
            const int row = tid >> 3, col = (tid & 7) * 8;
            const float* src = act + (size_t)(r0 + row) * ISH + kb + col;
            float* dst = As + row * LDA + col;
            *(v4f*)(dst)     = *(const v4f*)(src);
            *(v4f*)(dst + 4) = *(const v4f*)(src + 4);
        }
        stage_B(Bs, p0, p1, kb, ldb, tid);
        __syncthreads();
        #pragma unroll
        for (int kk = 0; kk < 64; kk += 4) {
            v2f a = *(const v2f*)(As + (16 * mrow + lr) * LDA + kk + 2 * hs);
            v2f b0 = B_FRAG(Bs, kk, hs, nb);
            ca = WMMA_F32(a, b0, ca);
            v2f b1 = B_FRAG(Bs, kk, hs, nb + 64);
            cb = WMMA_F32(a, b1, cb);
        }
    }
    #pragma unroll
    for (int g = 0; g < 8; ++g) {
        const int row = r0 + 16 * mrow + g + 8 * hs;
        const float s = tg[row];
        out[(size_t)row * HDIM + c0 + nb]      = ca[g] * s;
        out[(size_t)row * HDIM + c0 + 64 + nb] = cb[g] * s;
    }
}

// ---------------------------------------------------------------- expert gate-up + SwiGLU (gathered rows)
__global__ void expert_gateup_kernel(const float* __restrict__ x,     // [T,H]
                                     const float* __restrict__ wgu,   // [E,H,2I]
                                     const int*   __restrict__ counts,
                                     const int*   __restrict__ off,   // [E+1] padded(32)
                                     const int*   __restrict__ tok_list,
                                     float* __restrict__ act)         // [RPAD, I]
{
    __shared__ float As[32 * LDA];
    __shared__ float Bs[32 * LDBI];
    __shared__ int   toks[32];
    const int r0 = blockIdx.y * 32;
    if (r0 >= off[NEXP]) return;
    int e = 0;
    while (off[e + 1] <= r0) ++e;
    const int l0  = r0 - off[e];
    const int cnt = counts[e];

    const int tid  = threadIdx.x;
    const int lane = tid & 31;
    const int wv   = tid >> 5;
    const int mrow = wv >> 2;
    const int ncol = wv & 3;
    const int hs   = lane >> 4;
    const int lr   = lane & 15;
    const int nb   = 16 * ncol + lr;
    if (tid < 32) {
        const int li = l0 + tid;
        toks[tid] = (li < cnt) ? tok_list[e * CAPE + li] : -1;
    }

    const float* B   = wgu + (size_t)e * HDIM * (2 * IDIM);
    const int    ldb = 2 * IDIM;
    const int    c0g = blockIdx.x * 64;     // 0..IDIM-64
    const float* p0  = B + c0g;             // gate cols
    const float* p1  = B + IDIM + c0g;      // up cols
    v8f cg = {}; v8f cu = {};

    for (int kb = 0; kb < HDIM; kb += 64) {
        __syncthreads();
        {
            const int row = tid >> 3, col = (tid & 7) * 8;
            const int tk  = toks[row];
            float* dst = As + row * LDA + col;
            if (tk >= 0) {
                const float* src = x + (size_t)tk * HDIM + kb + col;
                *(v4f*)(dst)     = *(const v4f*)(src);
                *(v4f*)(dst + 4) = *(const v4f*)(src + 4);
            } else {
                v4f z = {};
                *(v4f*)(dst) = z; *(v4f*)(dst + 4) = z;
            }
        }
        stage_B(Bs, p0, p1, kb, ldb, tid);
        __syncthreads();
        #pragma unroll
        for (int kk = 0; kk < 64; kk += 4) {
            v2f a = *(const v2f*)(As + (16 * mrow + lr) * LDA + kk + 2 * hs);
            v2f b0 = B_FRAG(Bs, kk, hs, nb);
            cg = WMMA_F32(a, b0, cg);
            v2f b1 = B_FRAG(Bs, kk, hs, nb + 64);
            cu = WMMA_F32(a, b1, cu);
        }
    }
    #pragma unroll
    for (int g = 0; g < 8; ++g) {
        const int row = r0 + 16 * mrow + g + 8 * hs;     // padded row index
        act[(size_t)row * IDIM + c0g + nb] = silu_(cg[g]) * cu[g];
    }
}

// ---------------------------------------------------------------- expert down + weighted scatter-add
__global__ void expert_down_kernel(const float* __restrict__ act,    // [RPAD, I]
                                   const float* __restrict__ wd,     // [E,I,H]
                                   const int*   __restrict__ counts,
                                   const int*   __restrict__ off,
                                   const int*   __restrict__ tok_list,
                                   const float* __restrict__ w_list,
                                   float* __restrict__ out)          // [T,H]
{
    __shared__ float As[32 * LDA];
    __shared__ float Bs[32 * LDBI];
    const int r0 = blockIdx.y * 32;
    if (r0 >= off[NEXP]) return;
    int e = 0;
    while (off[e + 1] <= r0) ++e;
    const int l0  = r0 - off[e];
    const int cnt = counts[e];

    const int tid  = threadIdx.x;
    const int lane = tid & 31;
    const int wv   = tid >> 5;
    const int mrow = wv >> 2;
    const int ncol = wv & 3;
    const int hs   = lane >> 4;
    const int lr   = lane & 15;
    const int nb   = 16 * ncol + lr;

    const float* B   = wd + (size_t)e * IDIM * HDIM;
    const int    ldb = HDIM;
    const int    c0  = blockIdx.x * 128;
    const float* p0  = B + c0;
    const float* p1  = B + c0 + 64;
    v8f ca = {}; v8f cb = {};

    for (int kb = 0; kb < IDIM; kb += 64) {
        __syncthreads();
        {
            const int row = tid >> 3, col = (tid & 7) * 8;
            const float* src = act + (size_t)(r0 + row) * IDIM + kb + col;
            float* dst = As + row * LDA + col;
            *(v4f*)(dst)     = *(const v4f*)(src);
            *(v4f*)(dst + 4) = *(const v4f*)(src + 4);
        }
        stage_B(Bs, p0, p1, kb, ldb, tid);
        __syncthreads();
        #pragma unroll
        for (int kk = 0; kk < 64; kk += 4) {
            v2f a = *(const v2f*)(As + (16 * mrow + lr) * LDA + kk + 2 * hs);
            v2f b0 = B_FRAG(Bs, kk, hs, nb);
            ca = WMMA_F32(a, b0, ca);
            v2f b1 = B_FRAG(Bs, kk, hs, nb + 64);
            cb = WMMA_F32(a, b1, cb);
        }
    }
    #pragma unroll
    for (int g = 0; g < 8; ++g) {
        const int li = l0 + 16 * mrow + g + 8 * hs;
        if (li < cnt) {
            const int   tk  = tok_list[e * CAPE + li];
            const float wgt = w_list [e * CAPE + li];
            atomicAdd(&out[(size_t)tk * HDIM + c0 + nb],      wgt * ca[g]);
            atomicAdd(&out[(size_t)tk * HDIM + c0 + 64 + nb], wgt * cb[g]);
        }
    }
}

// ---------------------------------------------------------------- launch
extern "C" void kernel_launch(void* const* d_in, const int* in_sizes, int n_in,
                              void* d_out, int out_size, void* d_ws, size_t ws_size,
                              hipStream_t stream)
{
    (void)in_sizes; (void)n_in; (void)out_size; (void)ws_size;
    const float* x    = (const float*)d_in[0];   // [2,1024,1024]
    const float* gw   = (const float*)d_in[1];   // [1024,64]
    const float* sgw  = (const float*)d_in[2];   // [1024,1]
    const float* sguw = (const float*)d_in[3];   // [1024,4096]
    const float* sdw  = (const float*)d_in[4];   // [2048,1024]
    const float* eguw = (const float*)d_in[5];   // [64,1024,1024]
    const float* edw  = (const float*)d_in[6];   // [64,512,1024]
    float* out = (float*)d_out;                  // [2,1024,1024]

    char* ws = (char*)d_ws;
    size_t cur = 0;
    auto take = [&](size_t bytes) -> void* {
        cur = (cur + 255) & ~(size_t)255;
        void* p = ws + cur;
        cur += bytes;
        return p;
    };
    float* token_gate = (float*)take((size_t)T_TOK * 4);
    int*   counts     = (int*)  take((size_t)NEXP * 4);
    int*   eoff       = (int*)  take((size_t)(NEXP + 1) * 4);
    int*   tok_list   = (int*)  take((size_t)NEXP * CAPE * 4);
    float* w_list     = (float*)take((size_t)NEXP * CAPE * 4);
    float* act_sh     = (float*)take((size_t)T_TOK * ISH * 4);
    float* act_ex     = (float*)take((size_t)RPAD_MAX * IDIM * 4);

    hipMemsetAsync(counts, 0, NEXP * sizeof(int), stream);
    router_kernel<<<T_TOK, 64, 0, stream>>>(x, gw, sgw, token_gate, counts, tok_list, w_list);
    offsets_kernel<<<1, 32, 0, stream>>>(counts, eoff);
    shared_gateup_kernel<<<dim3(ISH / 64, T_TOK / 32), 256, 0, stream>>>(x, sguw, act_sh);
    shared_down_kernel<<<dim3(HDIM / 128, T_TOK / 32), 256, 0, stream>>>(act_sh, sdw, token_gate, out);
    expert_gateup_kernel<<<dim3(IDIM / 64, ROW_TILES), 256, 0, stream>>>(x, eguw, counts, eoff, tok_list, act_ex);
    expert_down_kernel<<<dim3(HDIM / 128, ROW_TILES), 256, 0, stream>>>(act_ex, edw, counts, eoff, tok_list, w_list, out);
}